// GatedDIP_24824910971079
// MI455X (gfx1250) — compile-verified
//
#include <hip/hip_runtime.h>
#include <math.h>

// GatedDIP on MI455X (gfx1250, wave32).
// - WMMA f32 16x16x4 for the latent->params GEMM (16x22x256, N padded to 32),
//   operands staged in LDS so the K-loop is ds_load_b64 + v_wmma only.
// - All bulk passes use float4 (global_load_b128) — workload is bandwidth-bound,
//   x/blur/out are L2-resident (192MB L2).
// - blur-V uses a 28x256 LDS tile to avoid 13x re-reads.
// - All cross-block reductions use integer atomics (deterministic).

#define Bn 16
#define Cn 3
#define Hn 512
#define Wn 512
#define HWn (Hn*Wn)
#define Dn 256
#define NBINS 1024
#define NUMPX 262          // (512*512)/1000
#define KS 13
#define KH 6

typedef float v2f __attribute__((ext_vector_type(2)));
typedef float v4f __attribute__((ext_vector_type(4)));
typedef float v8f __attribute__((ext_vector_type(8)));

__device__ __forceinline__ int reflect_idx(int i, int n) {
  if (i < 0) i = -i;
  if (i >= n) i = 2*n - 2 - i;
  return i;
}

// Monotonic float <-> orderable-uint encoding for deterministic atomic min/max.
__device__ __forceinline__ unsigned encOrd(float f) {
  unsigned u = __float_as_uint(f);
  return (u & 0x80000000u) ? ~u : (u | 0x80000000u);
}
__device__ __forceinline__ float decOrd(unsigned v) {
  unsigned u = (v & 0x80000000u) ? (v & 0x7FFFFFFFu) : ~v;
  return __uint_as_float(u);
}

// ---------------- init: zero hist/sums, seed min/max, compute gaussian taps --
__global__ void initK(unsigned* __restrict__ hist, unsigned* __restrict__ sumsU,
                      unsigned* __restrict__ mm, float* __restrict__ taps) {
  int i = blockIdx.x*blockDim.x + threadIdx.x;
  if (i < Bn*NBINS)                       hist[i] = 0u;
  else if (i < Bn*NBINS + 128)            sumsU[i - Bn*NBINS] = 0u;
  else if (i < Bn*NBINS + 128 + 8)        mm[i - (Bn*NBINS+128)] = 0xFFFFFFFFu; // min accum
  else if (i < Bn*NBINS + 128 + 16)       mm[i - (Bn*NBINS+128)] = 0u;          // max accum
  else if (i == Bn*NBINS + 128 + 16) {
    float k[KS]; float s = 0.f;
    for (int j = 0; j < KS; j++) { float xv = (float)(j - KH)/2.55f; k[j] = expf(-0.5f*xv*xv); s += k[j]; }
    for (int j = 0; j < KS; j++) taps[j] = k[j]/s;
  }
}

// ---------------- latent GEMM via V_WMMA_F32_16X16X4_F32 + param transforms --
struct WPtrs { const float *gw,*wbw,*gaw,*shw,*dfw,*cow,*tow; };
struct BPtrs { const float *gb,*wbb,*gab,*shb,*dfb,*cob,*tob; };

__device__ __forceinline__ float wrow(int r, int k, const WPtrs& wp) {
  if (r < 7)   return wp.gw[r*Dn + k];
  if (r < 10)  return wp.wbw[(r-7)*Dn + k];
  if (r == 10) return wp.gaw[k];
  if (r == 11) return wp.shw[k];
  if (r == 12) return wp.dfw[k];
  if (r == 13) return wp.cow[k];
  if (r < 22)  return wp.tow[(r-14)*Dn + k];
  return 0.f;
}
__device__ __forceinline__ float brow(int r, const BPtrs& bp) {
  if (r < 7)   return bp.gb[r];
  if (r < 10)  return bp.wbb[r-7];
  if (r == 10) return bp.gab[0];
  if (r == 11) return bp.shb[0];
  if (r == 12) return bp.dfb[0];
  if (r == 13) return bp.cob[0];
  if (r < 22)  return bp.tob[r-14];
  return 0.f;
}

__global__ void paramsK(const float* __restrict__ latent, WPtrs wp, BPtrs bp,
                        float* __restrict__ params) {
  __shared__ float Wl[32*Dn];     // concatenated+padded weight rows (32x256)
  __shared__ float Al[16*Dn];     // latent (16x256)
  __shared__ float Y[16][32];
  int tid = threadIdx.x;          // 256 threads = 8 waves
  // one-time branchy gather into LDS (keeps the WMMA loop clean)
  for (int i = tid; i < 32*Dn; i += 256) Wl[i] = wrow(i >> 8, i & 255, wp);
  for (int i = tid; i < 16*Dn; i += 256) Al[i] = latent[i];
  __syncthreads();
  if (tid < 32) {                 // wave 0 only, EXEC all ones
    int m = tid & 15, khalf = tid >> 4;   // ISA 16x4 f32 layout: lane-half selects K pair
    v8f acc0 = {0,0,0,0,0,0,0,0}, acc1 = {0,0,0,0,0,0,0,0};
    for (int k = 0; k < Dn; k += 4) {
      v2f a  = *(const v2f*)&Al[m*Dn        + k + 2*khalf];
      v2f b0 = *(const v2f*)&Wl[m*Dn        + k + 2*khalf];
      v2f b1 = *(const v2f*)&Wl[(m + 16)*Dn + k + 2*khalf];
      acc0 = __builtin_amdgcn_wmma_f32_16x16x4_f32(false, a, false, b0, (short)0, acc0, false, false);
      acc1 = __builtin_amdgcn_wmma_f32_16x16x4_f32(false, a, false, b1, (short)0, acc1, false, false);
    }
    // C/D layout: VGPR v -> row 8*khalf+v, col = lane&15
    #pragma unroll
    for (int v = 0; v < 8; v++) {
      Y[8*khalf + v][m]      = acc0[v];
      Y[8*khalf + v][16 + m] = acc1[v];
    }
  }
  __syncthreads();
  if (tid < 16) {
    int b = tid;
    float y[22];
    for (int r = 0; r < 22; r++) y[r] = Y[b][r] + brow(r, bp);
    float* p = params + b*32;
    for (int i = 0; i < 7; i++) p[i] = (tanhf(y[i])*0.5f + 0.5f)*0.99f + 0.01f;  // gates
    float wbr[3];
    for (int c = 0; c < 3; c++) wbr[c] = expf((tanhf(y[7+c])*0.5f + 0.5f) - 0.5f);
    float sc = 1.f/(1e-5f + 0.27f*wbr[0] + 0.67f*wbr[1] + 0.06f*wbr[2]);
    for (int c = 0; c < 3; c++) p[7+c] = sc*wbr[c];                               // wb
    const float lg = 0.9162907318741551f;                                         // log(2.5)
    p[10] = expf((tanhf(y[10])*0.5f + 0.5f)*(2.f*lg) - lg);                       // gamma
    p[11] = (tanhf(y[11])*0.5f + 0.5f)*0.9f + 0.1f;                               // sharp y
    p[12] = (tanhf(y[12])*0.5f + 0.5f)*0.9f + 0.1f;                               // omega
    p[13] = tanhf(y[13]);                                                          // alpha
    float ts = 0.f;
    for (int i = 0; i < 8; i++) { float tcv = (tanhf(y[14+i])*0.5f + 0.5f)*1.5f + 0.5f; p[14+i] = tcv; ts += tcv; }
    p[22] = ts + 1e-30f;                                                           // tc sum
  }
}

// ---------------- separable gaussian blur (reflect padding) -----------------
__global__ void blurHK(const float* __restrict__ x, const float* __restrict__ taps,
                       float* __restrict__ tmp) {
  __shared__ float s[256 + 2*KH];
  __shared__ float tk[KS];
  int tile = blockIdx.x;
  int wt   = tile & 1;            // W/256 == 2
  int rest = tile >> 1;
  int h    = rest & (Hn - 1);
  rest >>= 9;                     // (b*C + c)
  size_t plane = (size_t)rest;
  const float* row = x + (plane*Hn + h)*Wn;
  int w0 = wt*256;
  if (threadIdx.x < KS) tk[threadIdx.x] = taps[threadIdx.x];
  for (int i = threadIdx.x; i < 256 + 2*KH; i += 256)
    s[i] = row[reflect_idx(w0 - KH + i, Wn)];
  __syncthreads();
  float acc = 0.f;
  #pragma unroll
  for (int j = 0; j < KS; j++) acc += tk[j]*s[threadIdx.x + j];
  tmp[(plane*Hn + h)*Wn + w0 + threadIdx.x] = acc;
}

// blur-V: 256w x 16h output tile per block, 28x256 input tile in LDS.
__global__ void blurVK(const float* __restrict__ tmp, const float* __restrict__ taps,
                       float* __restrict__ blur) {
  __shared__ float tile[KS - 1 + 16][256];   // 28 rows
  __shared__ float tk[KS];
  int t  = blockIdx.x;
  int wt = t & 1;  t >>= 1;                  // 2 w-tiles
  int ht = t & 31; t >>= 5;                  // 32 h-tiles
  size_t plane = (size_t)t;                  // b*C + c
  int w0 = wt*256, h0 = ht*16;
  if (threadIdx.x < KS) tk[threadIdx.x] = taps[threadIdx.x];
  const float* base = tmp + plane*HWn;
  // stage 28 rows x 64 quads
  for (int i = threadIdx.x; i < 28*64; i += 256) {
    int row = i >> 6, wq = i & 63;
    int hin = reflect_idx(h0 - KH + row, Hn);
    v4f v = *(const v4f*)&base[(size_t)hin*Wn + w0 + wq*4];
    *(v4f*)&tile[row][wq*4] = v;
  }
  __syncthreads();
  int tw = threadIdx.x & 63;                 // w-quad
  int th = threadIdx.x >> 6;                 // 0..3
  float* ob = blur + plane*HWn;
  #pragma unroll
  for (int o = 0; o < 4; o++) {
    int hl = th*4 + o;
    v4f acc = {0.f, 0.f, 0.f, 0.f};
    #pragma unroll
    for (int j = 0; j < KS; j++) {
      v4f tv = *(const v4f*)&tile[hl + j][tw*4];
      acc += tk[j]*tv;
    }
    *(v4f*)&ob[(size_t)(h0 + hl)*Wn + w0 + tw*4] = acc;
  }
}

// ---------------- atmospheric light: histogram -> threshold -> average ------
__global__ void histK(const float* __restrict__ x, unsigned* __restrict__ hist) {
  __shared__ unsigned lh[NBINS];
  int b = blockIdx.y;
  for (int i = threadIdx.x; i < NBINS; i += blockDim.x) lh[i] = 0u;
  __syncthreads();
  const float* xb = x + (size_t)b*Cn*HWn;
  for (int q = blockIdx.x*blockDim.x + threadIdx.x; q < HWn/4; q += gridDim.x*blockDim.x) {
    int px = q*4;
    v4f r  = *(const v4f*)&xb[px];
    v4f g  = *(const v4f*)&xb[px + HWn];
    v4f bl = *(const v4f*)&xb[px + 2*HWn];
    #pragma unroll
    for (int j = 0; j < 4; j++) {
      float d = fminf(fminf(r[j], g[j]), bl[j]);
      int bin = (int)(d * (float)NBINS);
      bin = max(0, min(NBINS - 1, bin));
      atomicAdd(&lh[bin], 1u);
    }
  }
  __syncthreads();
  for (int i = threadIdx.x; i < NBINS; i += blockDim.x)
    if (lh[i]) atomicAdd(&hist[b*NBINS + i], lh[i]);
}

__global__ void thrK(const unsigned* __restrict__ hist, float* __restrict__ aL) {
  int b = blockIdx.x;
  if (threadIdx.x == 0) {
    unsigned cum = 0; int bin = NBINS - 1;
    for (; bin >= 0; bin--) { cum += hist[b*NBINS + bin]; if (cum >= NUMPX) break; }
    if (bin < 0) bin = 0;
    aL[b*4 + 3] = (float)bin / (float)NBINS;   // threshold (bin lower edge)
  }
}

__global__ void accumK(const float* __restrict__ x, const float* __restrict__ aL,
                       unsigned long long* __restrict__ sums) {
  int b = blockIdx.y;
  float thr = aL[b*4 + 3];
  const float* xb = x + (size_t)b*Cn*HWn;
  unsigned long long s0 = 0, s1 = 0, s2 = 0, cnt = 0;
  for (int q = blockIdx.x*blockDim.x + threadIdx.x; q < HWn/4; q += gridDim.x*blockDim.x) {
    int px = q*4;
    v4f r  = *(const v4f*)&xb[px];
    v4f g  = *(const v4f*)&xb[px + HWn];
    v4f bl = *(const v4f*)&xb[px + 2*HWn];
    #pragma unroll
    for (int j = 0; j < 4; j++) {
      float d = fminf(fminf(r[j], g[j]), bl[j]);
      if (d >= thr) {              // fixed point 2^30: deterministic int atomics
        s0 += (unsigned long long)(r[j]  * 1073741824.0f);
        s1 += (unsigned long long)(g[j]  * 1073741824.0f);
        s2 += (unsigned long long)(bl[j] * 1073741824.0f);
        cnt++;
      }
    }
  }
  __shared__ unsigned long long ls[4];
  if (threadIdx.x == 0) { ls[0] = ls[1] = ls[2] = ls[3] = 0ull; }
  __syncthreads();
  atomicAdd(&ls[0], s0); atomicAdd(&ls[1], s1); atomicAdd(&ls[2], s2); atomicAdd(&ls[3], cnt);
  __syncthreads();
  if (threadIdx.x == 0) {
    atomicAdd(&sums[b*4 + 0], ls[0]); atomicAdd(&sums[b*4 + 1], ls[1]);
    atomicAdd(&sums[b*4 + 2], ls[2]); atomicAdd(&sums[b*4 + 3], ls[3]);
  }
}

__global__ void finalizeAK(const unsigned long long* __restrict__ sums, float* __restrict__ aL) {
  int b = threadIdx.x;
  if (b < Bn) {
    unsigned long long cnt = sums[b*4 + 3]; if (!cnt) cnt = 1;
    float inv = 1.0f/((float)cnt * 1073741824.0f);
    for (int c = 0; c < 3; c++) aL[b*4 + c] = fmaxf((float)sums[b*4 + c]*inv, 1e-8f);
  }
}

// ---------------- fused per-pixel candidates (shared by reduce/combine) -----
__device__ __forceinline__ void computeCands(const float xv[3], const float bv[3],
    const float* __restrict__ P, const float* __restrict__ A3, float cand[3][6]) {
  float lum  = fminf(fmaxf(0.27f*xv[0] + 0.67f*xv[1] + 0.06f*xv[2], 0.f), 1.f);
  float clum = -cosf(3.14159265358979f*lum)*0.5f + 0.5f;
  float invl = clum/(lum + 1e-6f);
  float dark2 = fminf(fminf(xv[0]/A3[0], xv[1]/A3[1]), xv[2]/A3[2]);
  float t    = fmaxf(1.f - P[12]*dark2, 0.01f);
  float invt = 1.f/t;
  float alpha = P[13];
  float inv8ts = 8.f/P[22];
  #pragma unroll
  for (int c = 0; c < 3; c++) {
    float xc = xv[c];
    cand[c][0] = P[7+c]*xc;                                  // white balance
    cand[c][1] = powf(fmaxf(xc, 1e-4f), P[10]);              // gamma
    cand[c][2] = xc + P[11]*(xc - bv[c]);                    // sharpen
    cand[c][3] = (xc - A3[c])*invt + A3[c];                  // defog
    cand[c][4] = (1.f - alpha)*xc + alpha*(xc*invl);         // contrast
    float tv = 0.f;
    #pragma unroll
    for (int i = 0; i < 8; i++) tv += fminf(fmaxf(xc - 0.125f*(float)i, 0.f), 0.125f)*P[14+i];
    cand[c][5] = tv*inv8ts;                                  // tone curve
  }
}

__global__ void reduceK(const float* __restrict__ x, const float* __restrict__ blur,
                        const float* __restrict__ params, const float* __restrict__ aL,
                        unsigned* __restrict__ mm) {
  __shared__ float P[32];
  __shared__ float A3[4];
  int b = blockIdx.y;
  if (threadIdx.x < 23) P[threadIdx.x]  = params[b*32 + threadIdx.x];
  if (threadIdx.x < 3)  A3[threadIdx.x] = aL[b*4 + threadIdx.x];
  __syncthreads();
  float mn[6], mx[6];
  #pragma unroll
  for (int k = 0; k < 6; k++) { mn[k] = 3.4e38f; mx[k] = -3.4e38f; }
  const float* xb = x    + (size_t)b*Cn*HWn;
  const float* bb = blur + (size_t)b*Cn*HWn;
  for (int q = blockIdx.x*blockDim.x + threadIdx.x; q < HWn/4; q += gridDim.x*blockDim.x) {
    int px = q*4;
    v4f xr = *(const v4f*)&xb[px], xg = *(const v4f*)&xb[px + HWn], xl = *(const v4f*)&xb[px + 2*HWn];
    v4f br = *(const v4f*)&bb[px], bg = *(const v4f*)&bb[px + HWn], bl = *(const v4f*)&bb[px + 2*HWn];
    #pragma unroll
    for (int j = 0; j < 4; j++) {
      float xv[3] = { xr[j], xg[j], xl[j] };
      float bv[3] = { br[j], bg[j], bl[j] };
      float cand[3][6];
      computeCands(xv, bv, P, A3, cand);
      #pragma unroll
      for (int c = 0; c < 3; c++)
        #pragma unroll
        for (int k = 0; k < 6; k++) { mn[k] = fminf(mn[k], cand[c][k]); mx[k] = fmaxf(mx[k], cand[c][k]); }
    }
  }
  __shared__ float smin[6][256], smax[6][256];
  int tid = threadIdx.x;
  #pragma unroll
  for (int k = 0; k < 6; k++) { smin[k][tid] = mn[k]; smax[k][tid] = mx[k]; }
  for (int s = 128; s > 0; s >>= 1) {
    __syncthreads();
    if (tid < s)
      #pragma unroll
      for (int k = 0; k < 6; k++) {
        smin[k][tid] = fminf(smin[k][tid], smin[k][tid + s]);
        smax[k][tid] = fmaxf(smax[k][tid], smax[k][tid + s]);
      }
  }
  if (tid == 0)
    #pragma unroll
    for (int k = 0; k < 6; k++) {
      atomicMin(&mm[k],     encOrd(smin[k][0]));
      atomicMax(&mm[8 + k], encOrd(smax[k][0]));
    }
}

__global__ void combineK(const float* __restrict__ x, const float* __restrict__ blur,
                         const float* __restrict__ params, const float* __restrict__ aL,
                         unsigned* __restrict__ mm, float* __restrict__ out) {
  __shared__ float P[32], A3[4], NMN[6], NIV[6];
  int b = blockIdx.y;
  if (threadIdx.x < 23) P[threadIdx.x]  = params[b*32 + threadIdx.x];
  if (threadIdx.x < 3)  A3[threadIdx.x] = aL[b*4 + threadIdx.x];
  if (threadIdx.x < 6) {
    float lo = decOrd(mm[threadIdx.x]);
    float hi = decOrd(mm[8 + threadIdx.x]);
    NMN[threadIdx.x] = lo;
    NIV[threadIdx.x] = 1.f/(hi - lo);
  }
  __syncthreads();
  const int gidx[6] = {0, 1, 3, 4, 5, 6};   // gate index per candidate
  float omn = 3.4e38f, omx = -3.4e38f;
  const float* xb = x    + (size_t)b*Cn*HWn;
  const float* bb = blur + (size_t)b*Cn*HWn;
  float*       ob = out  + (size_t)b*Cn*HWn;
  for (int q = blockIdx.x*blockDim.x + threadIdx.x; q < HWn/4; q += gridDim.x*blockDim.x) {
    int px = q*4;
    v4f xr = *(const v4f*)&xb[px], xg = *(const v4f*)&xb[px + HWn], xl = *(const v4f*)&xb[px + 2*HWn];
    v4f br = *(const v4f*)&bb[px], bg = *(const v4f*)&bb[px + HWn], bl = *(const v4f*)&bb[px + 2*HWn];
    v4f o0, o1, o2;
    #pragma unroll
    for (int j = 0; j < 4; j++) {
      float xv[3] = { xr[j], xg[j], xl[j] };
      float bv[3] = { br[j], bg[j], bl[j] };
      float cand[3][6];
      computeCands(xv, bv, P, A3, cand);
      float oc[3];
      #pragma unroll
      for (int c = 0; c < 3; c++) {
        float o = xv[c]*P[2];                 // identity * gate2
        #pragma unroll
        for (int k = 0; k < 6; k++) o += (cand[c][k] - NMN[k])*NIV[k]*P[gidx[k]];
        oc[c] = o;
        omn = fminf(omn, o); omx = fmaxf(omx, o);
      }
      o0[j] = oc[0]; o1[j] = oc[1]; o2[j] = oc[2];
    }
    *(v4f*)&ob[px]         = o0;
    *(v4f*)&ob[px + HWn]   = o1;
    *(v4f*)&ob[px + 2*HWn] = o2;
  }
  __shared__ float smn[256], smx[256];
  int tid = threadIdx.x;
  smn[tid] = omn; smx[tid] = omx;
  for (int s = 128; s > 0; s >>= 1) {
    __syncthreads();
    if (tid < s) { smn[tid] = fminf(smn[tid], smn[tid + s]); smx[tid] = fmaxf(smx[tid], smx[tid + s]); }
  }
  if (tid == 0) { atomicMin(&mm[6], encOrd(smn[0])); atomicMax(&mm[14], encOrd(smx[0])); }
}

__global__ void normK(float* __restrict__ out, const unsigned* __restrict__ mm) {
  size_t q = (size_t)blockIdx.x*blockDim.x + threadIdx.x;  // quad index
  float lo  = decOrd(mm[6]);
  float inv = 1.f/(decOrd(mm[14]) - lo);
  v4f v = *(const v4f*)&out[q*4];
  #pragma unroll
  for (int j = 0; j < 4; j++) v[j] = (v[j] - lo)*inv;
  *(v4f*)&out[q*4] = v;
}

// ---------------------------------------------------------------------------
extern "C" void kernel_launch(void* const* d_in, const int* in_sizes, int n_in,
                              void* d_out, int out_size, void* d_ws, size_t ws_size,
                              hipStream_t stream) {
  (void)in_sizes; (void)n_in; (void)out_size; (void)ws_size;
  const float* x      = (const float*)d_in[0];
  const float* latent = (const float*)d_in[1];
  WPtrs wp{ (const float*)d_in[2],  (const float*)d_in[4],  (const float*)d_in[6],
            (const float*)d_in[8],  (const float*)d_in[10], (const float*)d_in[12],
            (const float*)d_in[14] };
  BPtrs bp{ (const float*)d_in[3],  (const float*)d_in[5],  (const float*)d_in[7],
            (const float*)d_in[9],  (const float*)d_in[11], (const float*)d_in[13],
            (const float*)d_in[15] };

  // workspace layout (~50.4 MB): blur | params | a/thr | taps | hist | sums | minmax
  float* blur   = (float*)d_ws;                       // 12,582,912 floats
  float* params = blur + (size_t)Bn*Cn*HWn;           // 16*32
  float* aL     = params + Bn*32;                     // 16*4 (a.rgb, thr)
  float* taps   = aL + Bn*4;                          // 16 floats (13 used)
  unsigned* hist = (unsigned*)(taps + 16);            // 16*1024
  unsigned long long* sums = (unsigned long long*)(hist + Bn*NBINS);  // 16*4
  unsigned* mm  = (unsigned*)(sums + Bn*4);           // min[8], max[8]
  float* tmp = (float*)d_out;                         // d_out doubles as blur-H scratch

  initK     <<<65, 256, 0, stream>>>(hist, (unsigned*)sums, mm, taps);
  paramsK   <<<1, 256, 0, stream>>>(latent, wp, bp, params);
  blurHK    <<<Bn*Cn*Hn*(Wn/256), 256, 0, stream>>>(x, taps, tmp);
  blurVK    <<<(Wn/256)*(Hn/16)*Bn*Cn, 256, 0, stream>>>(tmp, taps, blur);
  histK     <<<dim3(64, Bn), 256, 0, stream>>>(x, hist);
  thrK      <<<Bn, 32, 0, stream>>>(hist, aL);
  accumK    <<<dim3(64, Bn), 256, 0, stream>>>(x, aL, sums);
  finalizeAK<<<1, 64, 0, stream>>>(sums, aL);
  reduceK   <<<dim3(64, Bn), 256, 0, stream>>>(x, blur, params, aL, mm);
  combineK  <<<dim3(64, Bn), 256, 0, stream>>>(x, blur, params, aL, mm, (float*)d_out);
  normK     <<<(Bn*Cn*HWn/4)/256, 256, 0, stream>>>((float*)d_out, mm);
}